// RobertaForMultiLabelSequenceClassification_1503238554089
// MI455X (gfx1250) — compile-verified
//
#include <hip/hip_runtime.h>

// Problem constants (from reference): B=64, S=512, N=24, H=768, C=6
#define BB   64
#define SS   512
#define NSEG 24
#define HD   768
#define NCLS 6
#define NPAD 16   // class dim padded to WMMA tile width

typedef float v2f __attribute__((ext_vector_type(2)));
typedef float v8f __attribute__((ext_vector_type(8)));

__device__ __constant__ float d_CONST_LAST[NCLS] = {0.1f, 0.1f, 0.1f, 0.3f, 0.8f, 0.01f};
__device__ __constant__ float d_CONST_MID[NCLS]  = {0.1f, 0.1f, 0.1f, 0.8f, 0.3f, 0.01f};

// ---------------------------------------------------------------------------
// Kernel 0: build zero-padded weights Wpad[16][768] (rows >= 6 are zero).
// Lets the WMMA inner loop use unpredicated loads (no EXEC divergence).
// ---------------------------------------------------------------------------
__global__ __launch_bounds__(256) void pad_weights_kernel(
    const float* __restrict__ W,       // [C, H]
    float*       __restrict__ Wpad)    // [NPAD, H]
{
    const int idx = blockIdx.x * 256 + threadIdx.x;   // 0 .. 16*768-1
    if (idx >= NPAD * HD) return;
    const int row = idx / HD;
    Wpad[idx] = (row < NCLS) ? W[idx] : 0.0f;         // rows <6: same flat index
}

// ---------------------------------------------------------------------------
// Kernel 1: segment mean pooling (the bandwidth-bound part: ~100 MB stream).
// One block per (batch, segment); segments are disjoint so each token row is
// read exactly once. 192 threads x float4 = one coalesced 3 KB row per token.
// ---------------------------------------------------------------------------
__global__ __launch_bounds__(192) void seg_mean_kernel(
    const float* __restrict__ hidden,   // [B, S, H]
    const int*   __restrict__ ends,     // [B, N]
    float*       __restrict__ segmean)  // [B*N, H] (workspace)
{
    const int seg = blockIdx.x;                 // 0 .. B*NSEG-1
    const int b   = seg / NSEG;
    const int j   = seg % NSEG;

    const int end   = ends[b * NSEG + j];
    const int start = (j == 0) ? 0 : (ends[b * NSEG + j - 1] + 1);
    const int len   = end - start + 1;          // >= 1 by construction

    const int t4 = threadIdx.x;                 // which float4 of H
    const float4* __restrict__ base =
        (const float4*)(hidden + ((size_t)b * SS + (size_t)start) * HD);

    float4 s = make_float4(0.f, 0.f, 0.f, 0.f);
    for (int t = 0; t < len; ++t) {
        if (t + 8 < len) {
            // gfx1250 global_prefetch_b8 (speculative, RT)
            __builtin_prefetch((const void*)&base[(size_t)(t + 8) * (HD / 4) + t4], 0, 0);
        }
        float4 v = base[(size_t)t * (HD / 4) + t4];
        s.x += v.x; s.y += v.y; s.z += v.z; s.w += v.w;
    }
    const float inv = 1.0f / (float)len;
    s.x *= inv; s.y *= inv; s.z *= inv; s.w *= inv;
    ((float4*)(segmean + (size_t)seg * HD))[t4] = s;
}

// ---------------------------------------------------------------------------
// Kernel 2: [1536,768] x [768,16] GEMM via V_WMMA_F32_16X16X4_F32, fused with
// bias + sigmoid + (end > 500) constant override. One wave per 16-row tile.
//
// VGPR layouts (ISA 7.12.2, 32-bit operands, wave32):
//   A 16x4 : lane%16 = M row; lanes 0-15 hold K=0(v0),1(v1); lanes 16-31 K=2,3
//   B 4x16 : lane%16 = N col; same K split across lane halves / vgprs
//   C/D    : 8 VGPRs; row M = vgpr + 8*(lane/16); col N = lane%16
// Inner loop: 2x global_load_b64 + 1x v_wmma, fully unpredicated.
// ---------------------------------------------------------------------------
__global__ __launch_bounds__(32) void seg_gemm_wmma_kernel(
    const float* __restrict__ segmean,  // [B*N, H]
    const float* __restrict__ Wpad,     // [NPAD, H] zero-padded
    const float* __restrict__ bias,     // [C]
    const int*   __restrict__ ends,     // [B, N]
    float*       __restrict__ out)      // [B, N, C]
{
    const int lane = threadIdx.x;       // 0..31
    const int half = lane >> 4;         // 0 or 1 (selects K pair)
    const int lid  = lane & 15;         // row (A) / col (B,D) within tile
    const int m0   = blockIdx.x * 16;   // first segment row of this tile

    const float* __restrict__ aptr = segmean + (size_t)(m0 + lid) * HD + half * 2;
    const float* __restrict__ bptr = Wpad    + (size_t)lid        * HD + half * 2;

    v8f acc = {0.f, 0.f, 0.f, 0.f, 0.f, 0.f, 0.f, 0.f};

    for (int k = 0; k < HD; k += 4) {
        float2 av = *(const float2*)(aptr + k);   // A[lane%16][k + half*2 +{0,1}]
        float2 bv = *(const float2*)(bptr + k);   // Wt[k + half*2 +{0,1}][lane%16]
        v2f a;  a.x  = av.x; a.y  = av.y;
        v2f bm; bm.x = bv.x; bm.y = bv.y;
        acc = __builtin_amdgcn_wmma_f32_16x16x4_f32(
            /*neg_a=*/false, a, /*neg_b=*/false, bm,
            /*c_mod=*/(short)0, acc, /*reuse_a=*/false, /*reuse_b=*/false);
    }

    // Epilogue: bias + sigmoid + (end > 500) constant override.
    if (lid < NCLS) {
        const int   n  = lid;           // class index
        const float bb = bias[n];
        const float cl = d_CONST_LAST[n];
        const float cm = d_CONST_MID[n];
        #pragma unroll
        for (int r = 0; r < 8; ++r) {
            const int m    = m0 + r + half * 8;   // segment row for acc[r]
            const int bidx = m / NSEG;
            const int jseg = m % NSEG;
            float x = acc[r] + bb;
            float p = 1.0f / (1.0f + __expf(-x));
            const int e = ends[bidx * NSEG + jseg];
            if (e > 500) p = (jseg == NSEG - 1) ? cl : cm;
            out[(size_t)m * NCLS + n] = p;
        }
    }
}

// ---------------------------------------------------------------------------
// Launcher (graph-capture safe: kernel launches on `stream` only).
// Inputs (setup_inputs order): last_hidden_state f32 [64,512,768],
// sentence_token i32 [64,24], W f32 [6,768], b f32 [6].
// Output: f32 [64,24,6].
// Workspace: segmean [1536,768] f32 (4.72 MB) + Wpad [16,768] f32 (48 KB).
// ---------------------------------------------------------------------------
extern "C" void kernel_launch(void* const* d_in, const int* in_sizes, int n_in,
                              void* d_out, int out_size, void* d_ws, size_t ws_size,
                              hipStream_t stream) {
    const float* hidden = (const float*)d_in[0];
    const int*   ends   = (const int*)d_in[1];
    const float* W      = (const float*)d_in[2];
    const float* bias   = (const float*)d_in[3];
    float*       out    = (float*)d_out;

    float* segmean = (float*)d_ws;                          // [B*NSEG, HD]
    float* Wpad    = segmean + (size_t)BB * NSEG * HD;      // [NPAD, HD]

    (void)in_sizes; (void)n_in; (void)out_size; (void)ws_size;

    pad_weights_kernel<<<(NPAD * HD + 255) / 256, 256, 0, stream>>>(W, Wpad);
    seg_mean_kernel<<<BB * NSEG, 192, 0, stream>>>(hidden, ends, segmean);
    seg_gemm_wmma_kernel<<<(BB * NSEG) / 16, 32, 0, stream>>>(segmean, Wpad, bias, ends, out);
}